// TrigramModuleVanilla_86114094285207
// MI455X (gfx1250) — compile-verified
//
#include <hip/hip_runtime.h>
#include <math.h>

// Problem constants (from the reference)
#define NUM_CLASSES 601
#define OUT_DIM     27
#define P_ELEMS     (NUM_CLASSES * OUT_DIM)      // 16227 floats
#define P_CHUNKS    ((P_ELEMS + 3) / 4)          // 4057 x 16B chunks (reads 1 pad float)
#define GATHER_BLOCKS  3072
#define GATHER_THREADS 256

#if __has_builtin(__builtin_amdgcn_global_load_async_to_lds_b128)
#define HAVE_ASYNC_LDS 1
#endif

// Native clang vector types (usable with amdgcn/nontemporal builtins)
typedef int   v4i __attribute__((ext_vector_type(4)));
typedef float v4f __attribute__((ext_vector_type(4)));
typedef __attribute__((address_space(1))) v4i* gptr_v4i;
typedef __attribute__((address_space(3))) v4i* lptr_v4i;

// ---------------------------------------------------------------------------
// Kernel 1: P[r, :] = softmax(W[r, :])  -- one wave32 per row (27 <= 32 lanes)
// Matches reference numerics: counts = exp(x); probs = counts / sum(counts).
// Also writes one pad float at P[P_ELEMS] so B128 staging never reads junk.
// ---------------------------------------------------------------------------
__global__ __launch_bounds__(256) void softmax_rows_kernel(
    const float* __restrict__ W, float* __restrict__ P)
{
    const unsigned gtid = blockIdx.x * blockDim.x + threadIdx.x;
    const unsigned row  = gtid >> 5;          // wave32: one row per wave
    const unsigned lane = threadIdx.x & 31u;

    if (gtid == 0) P[P_ELEMS] = 0.0f;         // pad element for b128 staging

    if (row >= NUM_CLASSES) return;

    float e = 0.0f;
    if (lane < OUT_DIM) e = expf(W[row * OUT_DIM + lane]);

    // Wave-wide sum over 32 lanes (lanes >= 27 contribute 0)
    float s = e;
    #pragma unroll
    for (int off = 16; off >= 1; off >>= 1)
        s += __shfl_xor(s, off, 32);

    if (lane < OUT_DIM) P[row * OUT_DIM + lane] = e / s;
}

// ---------------------------------------------------------------------------
// Kernel 2: out[i, :] = P[bigram_idx[i], :]
// Stage P (65 KB) into LDS with the CDNA5 async copy engine (ASYNCcnt-
// tracked global_load_async_to_lds_b128), then serve the gather from LDS.
// Output written as coalesced non-temporal float4 (global_store_b128 th:NT)
// so the 452 MB write stream doesn't evict the hot table / index lines.
// ---------------------------------------------------------------------------
__global__ __launch_bounds__(GATHER_THREADS) void gather_probs_kernel(
    const int*   __restrict__ bidx,
    const float* __restrict__ P,
    float*       __restrict__ out,
    unsigned vec_total)   // total output floats / 4
{
    __shared__ __align__(16) float pl[P_ELEMS + 1];   // 16228 floats = 64912 B

    // ---- Stage P -> LDS ----------------------------------------------------
#if defined(HAVE_ASYNC_LDS)
    for (unsigned c = threadIdx.x; c < P_CHUNKS; c += blockDim.x) {
        __builtin_amdgcn_global_load_async_to_lds_b128(
            (gptr_v4i)(P + 4u * c),
            (lptr_v4i)(pl + 4u * c),
            /*offset=*/0, /*cpol=*/0);
    }
#if __has_builtin(__builtin_amdgcn_s_wait_asynccnt)
    __builtin_amdgcn_s_wait_asynccnt(0);
#else
    asm volatile("s_wait_asynccnt 0" ::: "memory");
#endif
#else
    for (unsigned c = threadIdx.x; c < P_CHUNKS; c += blockDim.x) {
        reinterpret_cast<v4f*>(pl)[c] =
            reinterpret_cast<const v4f*>(P)[c];
    }
#endif
    __syncthreads();

    // ---- Gather loop: one float4 of output per iteration -------------------
    const unsigned stride = gridDim.x * blockDim.x;
    for (unsigned v = blockIdx.x * blockDim.x + threadIdx.x;
         v < vec_total; v += stride) {
        unsigned e = v * 4u;              // first output element of this vec4
        unsigned i = e / 27u;             // batch row (magic-number divide)
        unsigned j = e - i * 27u;         // column within row
        int r = bidx[i];                  // wave-broadcast load, L0 hot

        float vals[4];
        #pragma unroll
        for (int k = 0; k < 4; ++k) {
            vals[k] = pl[(unsigned)r * 27u + j];
            if (k < 3) {                  // advance; reload idx on row wrap
                ++j;
                if (j == 27u) { j = 0u; r = bidx[++i]; }
            }
        }
        v4f o;
        o.x = vals[0]; o.y = vals[1]; o.z = vals[2]; o.w = vals[3];
#if __has_builtin(__builtin_nontemporal_store)
        __builtin_nontemporal_store(o, reinterpret_cast<v4f*>(out + e));
#else
        *reinterpret_cast<v4f*>(out + e) = o;
#endif
    }
}

// ---------------------------------------------------------------------------
extern "C" void kernel_launch(void* const* d_in, const int* in_sizes, int n_in,
                              void* d_out, int out_size, void* d_ws, size_t ws_size,
                              hipStream_t stream)
{
    const int*   bidx = (const int*)  d_in[0];   // [BATCH] int32
    const float* W    = (const float*)d_in[1];   // [601, 27] float32
    float*       out  = (float*)d_out;           // [BATCH, 27] float32
    float*       P    = (float*)d_ws;            // 16228 floats of scratch

    const unsigned batch = (unsigned)in_sizes[0];

    // Kernel 1: 601 waves -> 76 blocks of 256 threads (8 waves each)
    const int rows_per_block = 256 / 32;
    const int blocks1 = (NUM_CLASSES + rows_per_block - 1) / rows_per_block;
    softmax_rows_kernel<<<blocks1, 256, 0, stream>>>(W, P);

    // Kernel 2: bandwidth-bound gather
    const unsigned total     = batch * 27u;      // 113,246,208 (< 2^31)
    const unsigned vec_total = total / 4u;       // divisible: batch % 4 == 0
    gather_probs_kernel<<<GATHER_BLOCKS, GATHER_THREADS, 0, stream>>>(
        bidx, P, out, vec_total);
}